// knnFlow_1872605741571
// MI455X (gfx1250) — compile-verified
//
#include <hip/hip_runtime.h>

typedef __attribute__((ext_vector_type(2))) float v2f;
typedef __attribute__((ext_vector_type(8))) float v8f;

#define NB    8        // batches
#define NPTS  8192     // N1 == N2
#define NTILE (NPTS / 16)

// One block = one batch slice: 256 threads = 8 waves, each wave owns a 16-row
// tile of pc1 and scans all of pc2[batch] (staged in LDS) with
// v_wmma_f32_16x16x4_f32, keeping a per-lane-slot running argmin.
__global__ __launch_bounds__(256) void knn_flow_wmma(
    const float* __restrict__ pc1,   // [B,3,N]
    const float* __restrict__ pc2,   // [B,3,N]
    float* __restrict__ out)         // [B,3,N]  flow = pc2[nn] - pc1
{
    extern __shared__ float s2[];    // NPTS * 4 floats: {x, y, z, |p|^2}

    const int batch = blockIdx.x >> 6;          // 64 blocks per batch
    const int tgrp  = blockIdx.x & 63;          // 128 pc1 rows per block
    const int tid   = threadIdx.x;

    const float* p2b = pc2 + (size_t)batch * 3 * NPTS;
    const float* p1b = pc1 + (size_t)batch * 3 * NPTS;

    // ---- cooperative stage of pc2[batch] into LDS as float4 {x,y,z,sq} ----
    for (int i = tid; i < NPTS; i += 256) {
        float x = p2b[0 * NPTS + i];
        float y = p2b[1 * NPTS + i];
        float z = p2b[2 * NPTS + i];
        float sq = x * x + y * y + z * z;
        float4 v; v.x = x; v.y = y; v.z = z; v.w = sq;
        *(float4*)&s2[i * 4] = v;               // ds_store_b128
    }
    __syncthreads();

    const int wave = tid >> 5;
    const int lane = tid & 31;
    const int m    = lane & 15;                 // row/col within 16-tile
    const int half = lane >> 4;                 // K-half select (ISA 7.12.2)
    const int row_base = tgrp * 128 + wave * 16;
    const int row      = row_base + m;

    // ---- A fragment: row m = [-2x1, -2y1, -2z1, 1]
    // lanes 0-15: VGPR0=K0, VGPR1=K1 ; lanes 16-31: VGPR0=K2, VGPR1=K3
    v2f afrag;
    if (half == 0) {
        afrag.x = -2.0f * p1b[0 * NPTS + row];
        afrag.y = -2.0f * p1b[1 * NPTS + row];
    } else {
        afrag.x = -2.0f * p1b[2 * NPTS + row];
        afrag.y = 1.0f;
    }

    // B fragment lane address: col n = m, lane-half picks {x,y} or {z,sq}
    const float* sb = s2 + m * 4 + half * 2;

    float best[8];
    int   bidx[8];
    #pragma unroll
    for (int v = 0; v < 8; ++v) { best[v] = __builtin_inff(); bidx[v] = 0; }

    int n = m;                                  // this lane's current column id
    #pragma unroll 4
    for (int t = 0; t < NTILE; ++t) {
        v2f bfrag = *(const v2f*)(sb + t * 64); // ds_load_b64
        v8f c = {};
        // score[m][n] = |p2|^2 - 2<p1,p2>   (argmin-equivalent to distance)
        c = __builtin_amdgcn_wmma_f32_16x16x4_f32(
                /*neg_a=*/false, afrag, /*neg_b=*/false, bfrag,
                /*c_mod=*/(short)0, c, /*reuse_a=*/false, /*reuse_b=*/false);
        #pragma unroll
        for (int v = 0; v < 8; ++v) {
            float s  = c[v];
            bool  lt = s < best[v];
            best[v] = lt ? s : best[v];
            bidx[v] = lt ? n : bidx[v];
        }
        n += 16;
    }

    // ---- argmin reduce across the 16 column-lanes of each half-wave.
    // xor masks 8,4,2,1 never cross the 16-lane half boundary on wave32.
    #pragma unroll
    for (int v = 0; v < 8; ++v) {
        float val = best[v];
        int   idx = bidx[v];
        #pragma unroll
        for (int mask = 8; mask >= 1; mask >>= 1) {
            float oval = __shfl_xor(val, mask, 32);
            int   oidx = __shfl_xor(idx, mask, 32);
            bool take = (oval < val) || (oval == val && oidx < idx); // first-hit tie rule
            val = take ? oval : val;
            idx = take ? oidx : idx;
        }
        // slot v holds row row_base+v (lanes 0-15) / row_base+8+v (lanes 16-31)
        if (m == v) {
            int r = row_base + v + half * 8;
            float4 p = *(const float4*)&s2[idx * 4];
            out[(size_t)(batch * 3 + 0) * NPTS + r] = p.x - p1b[0 * NPTS + r];
            out[(size_t)(batch * 3 + 1) * NPTS + r] = p.y - p1b[1 * NPTS + r];
            out[(size_t)(batch * 3 + 2) * NPTS + r] = p.z - p1b[2 * NPTS + r];
        }
    }
}

extern "C" void kernel_launch(void* const* d_in, const int* in_sizes, int n_in,
                              void* d_out, int out_size, void* d_ws, size_t ws_size,
                              hipStream_t stream) {
    const float* pc1 = (const float*)d_in[0];
    const float* pc2 = (const float*)d_in[1];
    float* out = (float*)d_out;

    dim3 grid(NB * (NPTS / 128));   // 512 blocks: 8 batches x 64 row-groups
    dim3 block(256);                // 8 waves of 32
    size_t shmem = (size_t)NPTS * 4 * sizeof(float);  // 128 KB of 320 KB WGP LDS

    hipLaunchKernelGGL(knn_flow_wmma, grid, block, shmem, stream, pc1, pc2, out);
}